// Time2Vec_76845554860567
// MI455X (gfx1250) — compile-verified
//
#include <hip/hip_runtime.h>

typedef __attribute__((ext_vector_type(2))) float v2f;
typedef __attribute__((ext_vector_type(8))) float v8f;

// Time2Vec: out[token, d*64 + e] = (e==0) ? a : sin(a),  a = x[token,d]*W[d,e] + b[d,e]
// B=32, L=4096, D=8, E=64  ->  ntok = 131072 tokens, 512 outputs/token (256 MB written).
//
// Tiling: one wave32 per (16-token group, d, 16-wide e tile).
//   D = A(16x4 f32) x B(4x16 f32) + C  via V_WMMA_F32_16X16X4_F32:
//     A: K=0 column = x[token, d]  (lanes 0-15, VGPR0);  K=1..3 = 0
//     B: K=0 row    = W[d, e0+n]   (lanes 0-15, VGPR0);  rest   = 0
//     C: bias broadcast across M (all 8 acc VGPRs = b[d, e0+n])
//   Epilogue: v_sin_f32 everywhere except column e==0, then coalesced b32 stores.

__global__ __launch_bounds__(256) void t2v_wmma_kernel(
    const float* __restrict__ x,
    const float* __restrict__ W,
    const float* __restrict__ bias,
    float* __restrict__ out,
    int ntok)
{
  const int lane = threadIdx.x & 31;
  const int n    = lane & 15;   // N coordinate inside 16x16 tile
  const int half = lane >> 4;   // 0: lanes 0-15, 1: lanes 16-31

  const long long tokgroups = ((long long)ntok + 15) >> 4;
  const long long ntiles    = tokgroups * 32;  // 8 d-values * 4 e-tiles per token group
  const long long wave0     = (long long)blockIdx.x * (blockDim.x >> 5) + (threadIdx.x >> 5);
  const long long nwaves    = (long long)gridDim.x * (blockDim.x >> 5);

  for (long long tile = wave0; tile < ntiles; tile += nwaves) {
    const long long tg  = tile >> 5;
    const int       sub = (int)(tile & 31);
    const int       d   = sub >> 2;          // 0..7
    const int       e0  = (sub & 3) << 4;    // 0,16,32,48

    // x value for token tg*16 + n (every lane loads; only lanes 0-15 feed A's K=0 col)
    long long tokn = tg * 16 + n;
    if (tokn >= ntok) tokn = (long long)ntok - 1;
    const float xv = x[tokn * 8 + d];

    const int   col = d * 64 + e0 + n;       // output column in [0,512)
    const float wv  = W[col];
    const float bv  = bias[col];

    v2f a;  a.x  = (half == 0) ? xv : 0.0f;  a.y  = 0.0f;
    v2f bm; bm.x = (half == 0) ? wv : 0.0f;  bm.y = 0.0f;
    v8f c;
    #pragma unroll
    for (int r = 0; r < 8; ++r) c[r] = bv;

    v8f dacc;
#if __has_builtin(__builtin_amdgcn_wmma_f32_16x16x4_f32)
    // (neg_a, A, neg_b, B, c_mod, C, reuse_a, reuse_b)
    dacc = __builtin_amdgcn_wmma_f32_16x16x4_f32(false, a, false, bm,
                                                 (short)0, c, false, false);
#else
    // VALU fallback: x for token tg*16+m lives in lane m (m = 0..15)
    #pragma unroll
    for (int r = 0; r < 8; ++r) {
      const int   m  = r + half * 8;
      const float xm = __shfl(xv, m, 32);
      dacc[r] = fmaf(xm, wv, bv);
    }
#endif

    // Epilogue: raw value only for e == 0 (e0==0 && n==0), sin elsewhere.
    const bool keep = (e0 == 0) && (n == 0);
    #pragma unroll
    for (int r = 0; r < 8; ++r) {
      const long long token = tg * 16 + r + half * 8;  // M = r + 8*half
      float v = dacc[r];
      const float s = __sinf(v);   // v_sin_f32 (with 1/2pi premultiply)
      v = keep ? v : s;
      if (token < (long long)ntok)
        out[token * 512 + col] = v;
    }
  }
}

extern "C" void kernel_launch(void* const* d_in, const int* in_sizes, int n_in,
                              void* d_out, int out_size, void* d_ws, size_t ws_size,
                              hipStream_t stream) {
  const float* x = (const float*)d_in[0];   // (32,4096,8)  f32
  const float* W = (const float*)d_in[1];   // (8,64)       f32
  const float* b = (const float*)d_in[2];   // (8,64)       f32
  float* out = (float*)d_out;               // (32,4096,512) f32

  const int ntok = in_sizes[0] / 8;                      // B*L = 131072
  const long long tokgroups = ((long long)ntok + 15) / 16;
  const long long ntiles    = tokgroups * 32;            // 262144 wave-tiles
  // 8 waves per 256-thread block, ~8 tiles per wave -> ~4096 blocks
  long long waves  = (ntiles + 7) / 8;
  long long blocks = (waves + 7) / 8;
  if (blocks < 1) blocks = 1;
  if (blocks > (1LL << 24)) blocks = (1LL << 24);

  t2v_wmma_kernel<<<(dim3)(unsigned)blocks, 256, 0, stream>>>(x, W, b, out, ntok);
}